// Product_27994596836269
// MI455X (gfx1250) — compile-verified
//
#include <hip/hip_runtime.h>
#include <cstdint>

// Product layer forward, reduction='sum', cardinality=3, pad=1.
// x:   [256, 4, 512, 10, 8] f32  (inner 80 floats contiguous per feature)
// out: [256, 4, 171, 10, 8] f32
// out[nw, g, :] = x[nw, 3g, :] + x[nw, 3g+1, :] + (3g+2 < 512 ? x[nw, 3g+2, :] : 0)
//
// Bandwidth-bound: 168 MB in + 56 MB out, ~9.6 us floor at 23.3 TB/s.
// Strategy: async global->LDS staging (perfectly linear b128 loads on the
// ASYNCcnt-tracked copy path), 3:1 strided reduction out of LDS, b128 stores.

constexpr int kThreads       = 256;  // 8 wave32
constexpr int kInner4        = 20;   // 80 floats = 20 float4 per feature row
constexpr int kDIn           = 512;
constexpr int kDOut          = 171;
constexpr int kGroupsPerTile = 64;
constexpr int kFeatsPerTile  = 3 * kGroupsPerTile;           // 192
constexpr int kTiles         = 3;                            // ceil(171/64)
constexpr int kNW            = 256 * 4;                      // batch * weight_sets
constexpr int kFullN4        = kFeatsPerTile * kInner4;      // 3840 float4 staged

__device__ __forceinline__ void async_ld_b128(uint32_t lds_addr, uint64_t gaddr) {
    asm volatile("global_load_async_to_lds_b128 %0, %1, off"
                 :: "v"(lds_addr), "v"(gaddr)
                 : "memory");
}

__global__ __launch_bounds__(kThreads)
void product_sum3_kernel(const float* __restrict__ x, float* __restrict__ out)
{
    // 192 features * 20 float4 = 3840 float4 = 61440 bytes of LDS
    __shared__ float4 tile[kFeatsPerTile * kInner4];

    const int t   = blockIdx.x;              // feature tile within slice (0..2)
    const int nw  = blockIdx.y;              // which (n, w) slice (0..1023)
    const int g0  = t * kGroupsPerTile;      // first output group of tile
    const int f0  = 3 * g0;                  // first input feature of tile
    const int nf  = (kDIn  - f0 < kFeatsPerTile)  ? (kDIn  - f0) : kFeatsPerTile;  // 192,192,128
    const int ngr = (kDOut - g0 < kGroupsPerTile) ? (kDOut - g0) : kGroupsPerTile; // 64,64,43

    const float4* gin  = reinterpret_cast<const float4*>(x)
                       + (size_t)nw * (kDIn * kInner4) + (size_t)f0 * kInner4;
    float4*       gout = reinterpret_cast<float4*>(out)
                       + (size_t)nw * (kDOut * kInner4) + (size_t)g0 * kInner4;

    // addrspace(3) byte offset of the LDS tile: low 32 bits of the flat address
    // (flat LDS addresses are {shared_aperture, lds_offset[31:0]}).
    const uint32_t lds_base = (uint32_t)(uintptr_t)&tile[0];
    const int tid = threadIdx.x;

    // ---- Stage tile: async global -> LDS, 16B per lane per issue ----------
    // Full tiles: 15 back-to-back issues per lane, one wait. 32 lanes cover
    // 512B contiguous per issue -> fully linear HBM streams.
    if (nf == kFeatsPerTile) {
#pragma unroll
        for (int u = 0; u < kFullN4 / kThreads; ++u) {
            const int j = tid + u * kThreads;
            async_ld_b128(lds_base + (uint32_t)(j * 16),
                          (uint64_t)(uintptr_t)(gin + j));
        }
    } else {
        const int n4 = nf * kInner4;         // 2560 = 10 * 256 (last tile)
        for (int j = tid; j < n4; j += kThreads) {
            async_ld_b128(lds_base + (uint32_t)(j * 16),
                          (uint64_t)(uintptr_t)(gin + j));
        }
    }
    asm volatile("s_wait_asynccnt 0" ::: "memory");
    __syncthreads();

    // ---- Reduce groups of 3 features out of LDS, 128-bit stores -----------
    // Division-free (gl, i) walk: j += 256  <=>  gl += 12, i += 16 (mod 20).
    const int out4 = ngr * kInner4;
    int gl = tid / kInner4;                  // group within tile
    int i  = tid - gl * kInner4;             // float4 within the 80-float row
    for (int j = tid; j < out4; j += kThreads) {
        const int fl = 3 * gl;               // tile-local feature index

        float4 a = tile[(fl + 0) * kInner4 + i];
        float4 b = tile[(fl + 1) * kInner4 + i];
        float4 s;
        s.x = a.x + b.x; s.y = a.y + b.y; s.z = a.z + b.z; s.w = a.w + b.w;
        // Zero-pad handling: only global group 170 lacks its third term.
        if (f0 + fl + 2 < kDIn) {
            float4 c = tile[(fl + 2) * kInner4 + i];
            s.x += c.x; s.y += c.y; s.z += c.z; s.w += c.w;
        }
        gout[j] = s;

        gl += kThreads / kInner4;            // +12
        i  += kThreads % kInner4;            // +16
        if (i >= kInner4) { i -= kInner4; gl += 1; }
    }
}

extern "C" void kernel_launch(void* const* d_in, const int* in_sizes, int n_in,
                              void* d_out, int out_size, void* d_ws, size_t ws_size,
                              hipStream_t stream)
{
    (void)in_sizes; (void)n_in; (void)out_size; (void)d_ws; (void)ws_size;
    const float* x   = (const float*)d_in[0];
    float*       out = (float*)d_out;

    dim3 grid(kTiles, kNW);    // 3 feature tiles x 1024 (n,w) slices
    dim3 block(kThreads);
    hipLaunchKernelGGL(product_sum3_kernel, grid, block, 0, stream, x, out);
}